// Model_1_1_34153579938564
// MI455X (gfx1250) — compile-verified
//
#include <hip/hip_runtime.h>

typedef float v2f __attribute__((ext_vector_type(2)));
typedef float v8f __attribute__((ext_vector_type(8)));

#define SLOPE 0.01f

__device__ __forceinline__ float leaky(float x) { return x > 0.0f ? x : SLOPE * x; }

// ---------------- norm build ----------------
__global__ void fill1_k(float* p, int n) {
  int i = blockIdx.x * blockDim.x + threadIdx.x;
  if (i < n) p[i] = 1.0f;  // self-loop contribution to degree
}

__global__ void deg_scatter_k(const int* __restrict__ dst, const float* __restrict__ ew,
                              float* __restrict__ deg, int E) {
  int e = blockIdx.x * blockDim.x + threadIdx.x;
  if (e < E) atomicAdd(&deg[dst[e]], ew[e]);
}

__global__ void dis_self_k(const float* __restrict__ deg, float* __restrict__ dis,
                           float* __restrict__ selfnorm, int n) {
  int i = blockIdx.x * blockDim.x + threadIdx.x;
  if (i < n) {
    float d = deg[i];
    float r = d > 0.0f ? rsqrtf(fmaxf(d, 1e-30f)) : 0.0f;
    dis[i] = r;
    selfnorm[i] = r * r;  // norm of the self loop (weight 1)
  }
}

__global__ void edge_norm_k(const int* __restrict__ src, const int* __restrict__ dst,
                            const float* __restrict__ ew, const float* __restrict__ dis,
                            float* __restrict__ norm, int E) {
  int e = blockIdx.x * blockDim.x + threadIdx.x;
  if (e < E) norm[e] = dis[src[e]] * ew[e] * dis[dst[e]];
}

// ---------------- SpMM: y = A_norm * x ----------------
// 3-channel (layer 1) scalar versions
__global__ void selfinit3_k(const float* __restrict__ x, const float* __restrict__ selfnorm,
                            float* __restrict__ y, int n) {
  int idx = blockIdx.x * blockDim.x + threadIdx.x;
  if (idx < n * 3) {
    int node = idx / 3;
    y[idx] = selfnorm[node] * x[idx];
  }
}

__global__ void spmm3_k(const int* __restrict__ src, const int* __restrict__ dst,
                        const float* __restrict__ norm, const float* __restrict__ x,
                        float* __restrict__ y, int E) {
  int idx = blockIdx.x * blockDim.x + threadIdx.x;
  if (idx < E * 3) {
    int e = idx / 3;
    int c = idx - e * 3;
    atomicAdd(&y[dst[e] * 3 + c], norm[e] * x[src[e] * 3 + c]);
  }
}

// 64-channel: float4 self-loop init, float2-per-lane edge scatter (32 lanes/edge)
__global__ void selfinit64_k(const float4* __restrict__ x4, const float* __restrict__ selfnorm,
                             float4* __restrict__ y4, int n) {
  int idx = blockIdx.x * blockDim.x + threadIdx.x;  // n*16 float4s
  if (idx < n * 16) {
    int node = idx >> 4;
    float s = selfnorm[node];
    float4 v = x4[idx];
    v.x *= s; v.y *= s; v.z *= s; v.w *= s;
    y4[idx] = v;
  }
}

__global__ void spmm64_k(const int* __restrict__ src, const int* __restrict__ dst,
                         const float* __restrict__ norm, const float* __restrict__ x,
                         float* __restrict__ y, int E) {
  int idx = blockIdx.x * blockDim.x + threadIdx.x;  // E*32 lanes, 2 ch each
  if (idx < E * 32) {
    int e = idx >> 5;
    int c2 = (idx & 31) * 2;
    int s = src[e], d = dst[e];
    float nm = norm[e];
    float2 xv = *(const float2*)&x[s * 64 + c2];
    atomicAdd(&y[d * 64 + c2], nm * xv.x);
    atomicAdd(&y[d * 64 + c2 + 1], nm * xv.y);
  }
}

// ---------------- layer 1 dense: f0 = leaky(b + sum_k x_k @ w[k]) (CIN=3, H=64) ----------------
__global__ void layer1_dense_k(const float* __restrict__ x0, const float* __restrict__ x1,
                               const float* __restrict__ x2, const float* __restrict__ x3,
                               const float* __restrict__ w,  // [4,3,64]
                               const float* __restrict__ b, float* __restrict__ f0, int n) {
  int idx = blockIdx.x * blockDim.x + threadIdx.x;
  if (idx >= n * 64) return;
  int node = idx >> 6;
  int h = idx & 63;
  float acc = b[h];
#define TERM(K, XP)                                  \
  {                                                  \
    const float* xk = XP + node * 3;                 \
    const float* wk = w + (K * 3) * 64 + h;          \
    acc += xk[0] * wk[0] + xk[1] * wk[64] + xk[2] * wk[128]; \
  }
  TERM(0, x0) TERM(1, x1) TERM(2, x2) TERM(3, x3)
#undef TERM
  f0[idx] = leaky(acc);
}

// ---------------- bias init / leaky (float4) ----------------
__global__ void bias_init_k(float4* __restrict__ acc4, const float4* __restrict__ b4, int n) {
  int idx = blockIdx.x * blockDim.x + threadIdx.x;  // n*16 float4s
  if (idx < n * 16) acc4[idx] = b4[idx & 15];
}

__global__ void leaky_inplace_k(float4* __restrict__ p4, int total4) {
  int idx = blockIdx.x * blockDim.x + threadIdx.x;
  if (idx < total4) {
    float4 v = p4[idx];
    v.x = leaky(v.x); v.y = leaky(v.y); v.z = leaky(v.z); v.w = leaky(v.w);
    p4[idx] = v;
  }
}

// ---------------- WMMA f32 GEMM: Acc[N,64] += X[N,64] @ W[64,64] ----------------
// B operand staged in LDS as K-pair float2s: sW2[p][col] = (W[2p][col], W[2p+1][col]).
// A B fragment (K = k0,k0+1 with k0 even) is then a single contiguous ds_load_b64
// into an even-aligned VGPR pair -- no register shuffles between WMMAs.
// Pair-row stride 80 float2s => the two half-wave 128B runs are offset by
// 128B mod 256B => the wave's b64 read covers all 64 LDS banks exactly once.
// One wave computes a 16-row x 64-col strip; K=64 as 16 slices of K=4 via
// V_WMMA_F32_16X16X4_F32. EXEC all-1s at the WMMA (clamped loads, guarded stores).
#define WP2 80  // pair-row stride in float2
__global__ __launch_bounds__(256) void gemm64_acc_k(const float* __restrict__ X,
                                                    const float* __restrict__ W,
                                                    float* __restrict__ Acc, int nrows) {
  __shared__ float2 sW2[32 * WP2];  // 20 KB

  // stage W into pair layout: thread t covers col = t&63, pair-rows (t>>6)*8 .. +7
  {
    int col = threadIdx.x & 63;
    int p0 = (threadIdx.x >> 6) * 8;
#pragma unroll
    for (int j = 0; j < 8; ++j) {
      int pr = p0 + j;
      float2 v;
      v.x = W[(2 * pr) * 64 + col];      // coalesced across lanes
      v.y = W[(2 * pr + 1) * 64 + col];  // coalesced across lanes
      sW2[pr * WP2 + col] = v;
    }
  }
  __syncthreads();

  const int lane = threadIdx.x & 31;
  const int wave = threadIdx.x >> 5;
  const int tileRow = (blockIdx.x * 8 + wave) << 4;
  if (tileRow < nrows) {  // wave-uniform
    const int l16 = lane & 15;
    const int half = lane >> 4;

    // Load C (accumulate in place): lane l -> col = ct*16 + l%16, rows tileRow + j + 8*half
    v8f acc[4];
#pragma unroll
    for (int ct = 0; ct < 4; ++ct) {
      int col = ct * 16 + l16;
#pragma unroll
      for (int j = 0; j < 8; ++j) {
        int r = tileRow + j + 8 * half;
        int rc = r < nrows ? r : nrows - 1;
        acc[ct][j] = Acc[rc * 64 + col];
      }
    }

    int rowA = tileRow + l16;
    if (rowA >= nrows) rowA = nrows - 1;
    const float* xrow = X + rowA * 64;

#pragma unroll
    for (int kk = 0; kk < 16; ++kk) {
      int k0 = kk * 4 + half * 2;  // A/B layout: VGPR v, half h -> K = v + 2h (k0 even)
      int pr = k0 >> 1;
      v2f a;
      a.x = xrow[k0];
      a.y = xrow[k0 + 1];
#pragma unroll
      for (int ct = 0; ct < 4; ++ct) {
        int col = ct * 16 + l16;
        float2 w2 = sW2[pr * WP2 + col];  // single ds_load_b64
        v2f bf;
        bf.x = w2.x;
        bf.y = w2.y;
        acc[ct] = __builtin_amdgcn_wmma_f32_16x16x4_f32(false, a, false, bf, (short)0, acc[ct],
                                                        false, false);
      }
    }

#pragma unroll
    for (int ct = 0; ct < 4; ++ct) {
      int col = ct * 16 + l16;
#pragma unroll
      for (int j = 0; j < 8; ++j) {
        int r = tileRow + j + 8 * half;
        if (r < nrows) Acc[r * 64 + col] = acc[ct][j];
      }
    }
  }
}

// ---------------- pooling ----------------
__global__ void pool_zero_k(float* __restrict__ psum, float* __restrict__ pcnt, int G) {
  int idx = blockIdx.x * blockDim.x + threadIdx.x;
  if (idx < G * 64) psum[idx] = 0.0f;
  if (idx < G) pcnt[idx] = 0.0f;
}

// Batching is sorted: run-length accumulate PCHUNK nodes per thread, flush one
// atomic per (run, channel) -> ~16x fewer atomics than per-node scatter.
#define PCHUNK 16
__global__ void pool_scatter_k(const float* __restrict__ h, const int* __restrict__ batching,
                               float* __restrict__ psum, float* __restrict__ pcnt, int n) {
  int t = blockIdx.x * blockDim.x + threadIdx.x;
  int nchunks = (n + PCHUNK - 1) / PCHUNK;
  int chunk = t >> 6;
  int c = t & 63;
  if (chunk >= nchunks) return;
  int n0 = chunk * PCHUNK;
  int n1 = n0 + PCHUNK;
  if (n1 > n) n1 = n;
  int gcur = batching[n0];
  float s = 0.0f, cnt = 0.0f;
  for (int node = n0; node < n1; ++node) {
    int g = batching[node];
    if (g != gcur) {
      atomicAdd(&psum[gcur * 64 + c], s);
      if (c == 0) atomicAdd(&pcnt[gcur], cnt);
      gcur = g; s = 0.0f; cnt = 0.0f;
    }
    s += h[node * 64 + c];
    cnt += 1.0f;
  }
  atomicAdd(&psum[gcur * 64 + c], s);
  if (c == 0) atomicAdd(&pcnt[gcur], cnt);
}

// ---------------- MLP head: 64 -> 64 -> 64 -> 2, twice (phi/theta) ----------------
__global__ __launch_bounds__(64) void mlp_head_k(
    const float* __restrict__ psum, const float* __restrict__ pcnt,
    const float* __restrict__ pw1, const float* __restrict__ pb1,
    const float* __restrict__ pw2, const float* __restrict__ pb2,
    const float* __restrict__ pw3, const float* __restrict__ pb3,
    const float* __restrict__ tw1, const float* __restrict__ tb1,
    const float* __restrict__ tw2, const float* __restrict__ tb2,
    const float* __restrict__ tw3, const float* __restrict__ tb3, float* __restrict__ out) {
  __shared__ float s_pool[64], s_a[64], s_b[64];
  int g = blockIdx.x;
  int t = threadIdx.x;
  float cnt = fmaxf(pcnt[g], 1.0f);
  s_pool[t] = psum[g * 64 + t] / cnt;
  __syncthreads();

  float v = pb1[t];
  for (int c = 0; c < 64; ++c) v += s_pool[c] * pw1[c * 64 + t];
  s_a[t] = leaky(v);
  __syncthreads();
  v = pb2[t];
  for (int c = 0; c < 64; ++c) v += s_a[c] * pw2[c * 64 + t];
  s_b[t] = leaky(v);
  __syncthreads();
  if (t < 2) {
    float o = pb3[t];
    for (int c = 0; c < 64; ++c) o += s_b[c] * pw3[c * 2 + t];
    out[g * 4 + t] = o;
  }
  __syncthreads();

  v = tb1[t];
  for (int c = 0; c < 64; ++c) v += s_pool[c] * tw1[c * 64 + t];
  s_a[t] = leaky(v);
  __syncthreads();
  v = tb2[t];
  for (int c = 0; c < 64; ++c) v += s_a[c] * tw2[c * 64 + t];
  s_b[t] = leaky(v);
  __syncthreads();
  if (t < 2) {
    float o = tb3[t];
    for (int c = 0; c < 64; ++c) o += s_b[c] * tw3[c * 2 + t];
    out[g * 4 + 2 + t] = o;
  }
}

// ---------------- host side ----------------
static inline size_t alignup(size_t x) { return (x + 255) & ~(size_t)255; }

extern "C" void kernel_launch(void* const* d_in, const int* in_sizes, int n_in, void* d_out,
                              int out_size, void* d_ws, size_t ws_size, hipStream_t stream) {
  const float* X = (const float*)d_in[0];
  const int* EI = (const int*)d_in[1];
  const float* EW = (const float*)d_in[2];
  const int* Batching = (const int*)d_in[3];
  const float* c1_w = (const float*)d_in[4];
  const float* c1_b = (const float*)d_in[5];
  const float* c2_w = (const float*)d_in[6];
  const float* c2_b = (const float*)d_in[7];
  const float* pw1 = (const float*)d_in[8];
  const float* pb1 = (const float*)d_in[9];
  const float* pw2 = (const float*)d_in[10];
  const float* pb2 = (const float*)d_in[11];
  const float* pw3 = (const float*)d_in[12];
  const float* pb3 = (const float*)d_in[13];
  const float* tw1 = (const float*)d_in[14];
  const float* tb1 = (const float*)d_in[15];
  const float* tw2 = (const float*)d_in[16];
  const float* tb2 = (const float*)d_in[17];
  const float* tw3 = (const float*)d_in[18];
  const float* tb3 = (const float*)d_in[19];
  float* out = (float*)d_out;

  const int E = in_sizes[2];   // edges
  const int N = in_sizes[3];   // nodes
  const int G = out_size / 4;  // graphs

  const int* src = EI;
  const int* dst = EI + E;

  // carve workspace
  char* p = (char*)d_ws;
  float* deg = (float*)p;       p += alignup((size_t)N * 4);
  float* dis = (float*)p;       p += alignup((size_t)N * 4);
  float* selfnorm = (float*)p;  p += alignup((size_t)N * 4);
  float* norm = (float*)p;      p += alignup((size_t)E * 4);
  float* h1 = (float*)p;        p += alignup((size_t)N * 3 * 4);
  float* h2 = (float*)p;        p += alignup((size_t)N * 3 * 4);
  float* h3 = (float*)p;        p += alignup((size_t)N * 3 * 4);
  float* bufA = (float*)p;      p += alignup((size_t)N * 64 * 4);
  float* bufB = (float*)p;      p += alignup((size_t)N * 64 * 4);
  float* accb = (float*)p;      p += alignup((size_t)N * 64 * 4);
  float* psum = (float*)p;      p += alignup((size_t)G * 64 * 4);
  float* pcnt = (float*)p;      p += alignup((size_t)G * 4);
  (void)ws_size; (void)n_in;

  const int TB = 256;
  auto g1 = [&](long long t) { return (unsigned)((t + TB - 1) / TB); };

  // gcn_norm
  fill1_k<<<g1(N), TB, 0, stream>>>(deg, N);
  deg_scatter_k<<<g1(E), TB, 0, stream>>>(dst, EW, deg, E);
  dis_self_k<<<g1(N), TB, 0, stream>>>(deg, dis, selfnorm, N);
  edge_norm_k<<<g1(E), TB, 0, stream>>>(src, dst, EW, dis, norm, E);

  // layer-1 hops (3 channels)
  selfinit3_k<<<g1((long long)N * 3), TB, 0, stream>>>(X, selfnorm, h1, N);
  spmm3_k<<<g1((long long)E * 3), TB, 0, stream>>>(src, dst, norm, X, h1, E);
  selfinit3_k<<<g1((long long)N * 3), TB, 0, stream>>>(h1, selfnorm, h2, N);
  spmm3_k<<<g1((long long)E * 3), TB, 0, stream>>>(src, dst, norm, h1, h2, E);
  selfinit3_k<<<g1((long long)N * 3), TB, 0, stream>>>(h2, selfnorm, h3, N);
  spmm3_k<<<g1((long long)E * 3), TB, 0, stream>>>(src, dst, norm, h2, h3, E);

  // layer-1 dense + leaky -> bufA (f0)
  layer1_dense_k<<<g1((long long)N * 64), TB, 0, stream>>>(X, h1, h2, h3, c1_w, c1_b, bufA, N);

  // layer-2: acc = b + sum_k f_k @ w[k]
  const int gemmBlocks = (N + 127) / 128;  // 8 waves x 16 rows per block
  bias_init_k<<<g1((long long)N * 16), TB, 0, stream>>>((float4*)accb, (const float4*)c2_b, N);
  gemm64_acc_k<<<gemmBlocks, 256, 0, stream>>>(bufA, c2_w, accb, N);

  // f1 = A f0
  selfinit64_k<<<g1((long long)N * 16), TB, 0, stream>>>((const float4*)bufA, selfnorm,
                                                         (float4*)bufB, N);
  spmm64_k<<<g1((long long)E * 32), TB, 0, stream>>>(src, dst, norm, bufA, bufB, E);
  gemm64_acc_k<<<gemmBlocks, 256, 0, stream>>>(bufB, c2_w + 4096, accb, N);

  // f2 = A f1
  selfinit64_k<<<g1((long long)N * 16), TB, 0, stream>>>((const float4*)bufB, selfnorm,
                                                         (float4*)bufA, N);
  spmm64_k<<<g1((long long)E * 32), TB, 0, stream>>>(src, dst, norm, bufB, bufA, E);
  gemm64_acc_k<<<gemmBlocks, 256, 0, stream>>>(bufA, c2_w + 8192, accb, N);

  // f3 = A f2
  selfinit64_k<<<g1((long long)N * 16), TB, 0, stream>>>((const float4*)bufA, selfnorm,
                                                         (float4*)bufB, N);
  spmm64_k<<<g1((long long)E * 32), TB, 0, stream>>>(src, dst, norm, bufA, bufB, E);
  gemm64_acc_k<<<gemmBlocks, 256, 0, stream>>>(bufB, c2_w + 12288, accb, N);

  // leaky -> node embeddings h
  leaky_inplace_k<<<g1((long long)N * 16), TB, 0, stream>>>((float4*)accb, N * 16);

  // global mean pool (Batching sorted -> run-length accumulate)
  pool_zero_k<<<g1((long long)G * 64), TB, 0, stream>>>(psum, pcnt, G);
  {
    long long pthreads = (long long)((N + PCHUNK - 1) / PCHUNK) * 64;
    pool_scatter_k<<<g1(pthreads), TB, 0, stream>>>(accb, Batching, psum, pcnt, N);
  }

  // MLP heads -> out [G,4]
  mlp_head_k<<<G, 64, 0, stream>>>(psum, pcnt, pw1, pb1, pw2, pb2, pw3, pb3, tw1, tb1, tw2, tb2,
                                   tw3, tb3, out);
}